// OntoAlignEncoder_58875411694220
// MI455X (gfx1250) — compile-verified
//
#include <hip/hip_runtime.h>

typedef __attribute__((ext_vector_type(2))) float v2f;
typedef __attribute__((ext_vector_type(8))) float v8f;

// amdgcn builtins are only visible in the device pass; don't probe them on host.
#if defined(__HIP_DEVICE_COMPILE__) && !__has_builtin(__builtin_amdgcn_wmma_f32_16x16x4_f32)
#error "builtin __builtin_amdgcn_wmma_f32_16x16x4_f32 not available on this toolchain"
#endif

__device__ __forceinline__ v8f wmma_f32(v2f a, v2f b, v8f c) {
  // 8 args: (neg_a, A, neg_b, B, c_mod, C, reuse_a, reuse_b) -> v_wmma_f32_16x16x4_f32
  return __builtin_amdgcn_wmma_f32_16x16x4_f32(false, a, false, b, (short)0, c, false, false);
}

// C[nrows,128] = act(A[nrows,KTOT] @ W[KTOT,128] + bias), exact fp32 via WMMA.
// CONCAT: A = [X0 | X1], both [nrows,128] row-major (KTOT must be 256).
// act: 0=none 1=relu 2=sigmoid
template<int KTOT, bool CONCAT>
__global__ __launch_bounds__(256, 2) void gemm_kernel(
    const float* __restrict__ X0, const float* __restrict__ X1,
    const float* __restrict__ W, const float* __restrict__ bias,
    float* __restrict__ out, int nrows, int act)
{
  __shared__ float As[32][68];    // 32 rows x 64 k, pad 4 -> conflict-free fragment reads
  __shared__ float Ws[64][136];   // 64 k x 128 n, pad 8 -> lane halves hit disjoint banks

  const int tid  = threadIdx.x;
  const int rowbase = blockIdx.x * 32;
  const int wv = tid >> 5, lane = tid & 31;
  const int q  = lane & 15;          // M index (A) / N index (B)
  const int hi = lane >> 4;          // half-wave select
  const int kb = hi * 2;             // K sub-offset per ISA 16x4 layout
  const int rt = (wv >> 2) * 16;     // row tile base (0 or 16)
  const int c0 = (wv & 3) * 32;      // this wave's two column tiles
  const int c1 = c0 + 16;

  v8f acc0 = {}; v8f acc1 = {};

  for (int kc = 0; kc < KTOT; kc += 64) {
    // ---- cooperative load A chunk: 32x64 floats (512 float4, 2 per thread)
    #pragma unroll
    for (int i = 0; i < 2; ++i) {
      int idx = tid + i * 256;
      int r = idx >> 4, c4 = (idx & 15) * 4;
      int row = rowbase + r;
      float4 v = make_float4(0.f, 0.f, 0.f, 0.f);
      if (row < nrows) {
        int kg = kc + c4;
        const float* p;
        if (CONCAT) p = (kg < 128) ? (X0 + (size_t)row * 128 + kg)
                                   : (X1 + (size_t)row * 128 + (kg - 128));
        else        p = X0 + (size_t)row * KTOT + kg;
        v = *(const float4*)p;
      }
      *(float4*)&As[r][c4] = v;
    }
    // ---- cooperative load W chunk: 64x128 floats (2048 float4, 8 per thread)
    #pragma unroll
    for (int i = 0; i < 8; ++i) {
      int idx = tid + i * 256;
      int r = idx >> 5, c4 = (idx & 31) * 4;
      *(float4*)&Ws[r][c4] = *(const float4*)(W + (size_t)(kc + r) * 128 + c4);
    }
    if (kc + 64 < KTOT) __builtin_prefetch(W + (size_t)(kc + 64) * 128 + tid * 4, 0, 1);
    __syncthreads();

    #pragma unroll
    for (int k4 = 0; k4 < 16; ++k4) {
      const int k = k4 * 4 + kb;
      v2f a;  a.x  = As[rt + q][k];   a.y  = As[rt + q][k + 1];
      v2f b0; b0.x = Ws[k][c0 + q];   b0.y = Ws[k + 1][c0 + q];
      v2f b1; b1.x = Ws[k][c1 + q];   b1.y = Ws[k + 1][c1 + q];
      acc0 = wmma_f32(a, b0, acc0);
      acc1 = wmma_f32(a, b1, acc1);
    }
    __syncthreads();
  }

  // ---- epilogue: C/D layout -> VGPR v holds (M=v|v+8, N=q)
  float bc0 = bias ? bias[c0 + q] : 0.0f;
  float bc1 = bias ? bias[c1 + q] : 0.0f;
  #pragma unroll
  for (int v = 0; v < 8; ++v) {
    int row = rowbase + rt + v + hi * 8;
    if (row >= nrows) continue;
    float x0 = acc0[v] + bc0;
    float x1 = acc1[v] + bc1;
    if (act == 1) { x0 = fmaxf(x0, 0.f); x1 = fmaxf(x1, 0.f); }
    else if (act == 2) { x0 = 1.f / (1.f + __expf(-x0)); x1 = 1.f / (1.f + __expf(-x1)); }
    out[(size_t)row * 128 + c0 + q] = x0;
    out[(size_t)row * 128 + c1 + q] = x1;
  }
}

// h,pc *= domain_emb[domain_ids[batch[node]]]
__global__ void modulate_kernel(float* __restrict__ h, float* __restrict__ pc,
                                const float* __restrict__ de, const int* __restrict__ dom,
                                const int* __restrict__ batch, int total) {
  int i = blockIdx.x * blockDim.x + threadIdx.x;
  if (i >= total) return;
  int node = i >> 7, hh = i & 127;
  float dm = de[(size_t)dom[batch[node]] * 128 + hh];
  h[i] *= dm; pc[i] *= dm;
}

// per-(src,prop) counts + per-(dst,rel) counts in one edge pass (cheap atomics)
__global__ void edge_counts_kernel(const int* __restrict__ src, const int* __restrict__ dst,
                                   const int* __restrict__ et, const int* __restrict__ pid,
                                   float* __restrict__ cntp, float* __restrict__ cnt, int E) {
  int e = blockIdx.x * blockDim.x + threadIdx.x;
  if (e >= E) return;
  int p = pid[e]; p = p < 0 ? 0 : (p > 15 ? 15 : p);
  atomicAdd(&cntp[(size_t)src[e] * 16 + p], 1.0f);
  atomicAdd(&cnt[(size_t)dst[e] * 8 + et[e]], 1.0f);
}

// prop_agg[n] = (sum_p cntp[n,p] * pe[p]) / max(deg,1)   (deg = sum_p cntp[n,p])
__global__ void prop_kernel(const float* __restrict__ cntp, const float* __restrict__ pe,
                            float* __restrict__ prop) {
  __shared__ float c[16];
  int node = blockIdx.x;
  if (threadIdx.x < 16) c[threadIdx.x] = cntp[(size_t)node * 16 + threadIdx.x];
  __syncthreads();
  float deg = 0.f;
  #pragma unroll
  for (int p = 0; p < 16; ++p) deg += c[p];
  deg = fmaxf(deg, 1.0f);
  int hh = threadIdx.x;
  float s = 0.f;
  #pragma unroll
  for (int p = 0; p < 16; ++p) s += c[p] * pe[p * 128 + hh];
  prop[(size_t)node * 128 + hh] = s / deg;
}

// h = pg*h + (1-pg)*pc + prop
__global__ void combine1_kernel(float* __restrict__ h, const float* __restrict__ pc,
                                const float* __restrict__ pg, const float* __restrict__ prop,
                                int total) {
  int i = blockIdx.x * blockDim.x + threadIdx.x;
  if (i >= total) return;
  float g = pg[i];
  h[i] = g * h[i] + (1.0f - g) * pc[i] + prop[i];
}

__global__ void add_kernel(float* __restrict__ o, const float* __restrict__ a,
                           const float* __restrict__ b, int total) {
  int i = blockIdx.x * blockDim.x + threadIdx.x;
  if (i < total) o[i] = a[i] + b[i];
}

__global__ void relu_kernel(float* __restrict__ o, const float* __restrict__ a, int total) {
  int i = blockIdx.x * blockDim.x + threadIdx.x;
  if (i < total) o[i] = fmaxf(a[i], 0.0f);
}

// Weff = root + SCALING * lora_a @ lora_b   (128x128, rank 8, SCALING = 16/8 = 2)
__global__ void weff_kernel(const float* __restrict__ root, const float* __restrict__ la,
                            const float* __restrict__ lb, float* __restrict__ weff) {
  int idx = blockIdx.x * 256 + threadIdx.x;   // 16384 total
  int k = idx >> 7, d = idx & 127;
  float s = 0.f;
  #pragma unroll
  for (int j = 0; j < 8; ++j) s += la[k * 8 + j] * lb[j * 128 + d];
  weff[idx] = root[idx] + 2.0f * s;
}

// conv[dst] += zr[src] / max(cnt[dst,rel],1) for edges of `rel`; 8 edges/block, 32 lanes/edge
__global__ void scatter_rel_kernel(const float* __restrict__ zr, const int* __restrict__ src,
                                   const int* __restrict__ dst, const int* __restrict__ et,
                                   const float* __restrict__ cnt, float* __restrict__ conv,
                                   int E, int rel) {
  int e = blockIdx.x * 8 + (threadIdx.x >> 5);
  if (e >= E) return;
  if (et[e] != rel) return;
  int s = src[e], d = dst[e];
  float w = 1.0f / fmaxf(cnt[(size_t)d * 8 + rel], 1.0f);
  int lane = threadIdx.x & 31;
  float4 v = *(const float4*)(zr + (size_t)s * 128 + lane * 4);
  float* o = conv + (size_t)d * 128 + lane * 4;
  atomicAdd(o + 0, w * v.x);
  atomicAdd(o + 1, w * v.y);
  atomicAdd(o + 2, w * v.z);
  atomicAdd(o + 3, w * v.w);
}

// out = fg*sem + (1-fg)*stru
__global__ void final_kernel(float* __restrict__ out, const float* __restrict__ fg,
                             const float* __restrict__ sem, const float* __restrict__ stru,
                             int total) {
  int i = blockIdx.x * blockDim.x + threadIdx.x;
  if (i >= total) return;
  float g = fg[i];
  out[i] = g * sem[i] + (1.0f - g) * stru[i];
}

extern "C" void kernel_launch(void* const* d_in, const int* in_sizes, int n_in,
                              void* d_out, int out_size, void* d_ws, size_t ws_size,
                              hipStream_t stream) {
  (void)n_in; (void)out_size; (void)ws_size;
  const float* x_text   = (const float*)d_in[0];
  const float* x_path   = (const float*)d_in[1];
  const float* text_w   = (const float*)d_in[2];
  const float* text_b   = (const float*)d_in[3];
  const float* path_w   = (const float*)d_in[4];
  const float* path_b   = (const float*)d_in[5];
  const float* dom_emb  = (const float*)d_in[6];
  const float* prop_emb = (const float*)d_in[7];
  const float* fus_w    = (const float*)d_in[8];
  const float* fus_b    = (const float*)d_in[9];
  const float* pg1_w    = (const float*)d_in[10];
  const float* pg1_b    = (const float*)d_in[11];
  const float* pg2_w    = (const float*)d_in[12];
  const float* pg2_b    = (const float*)d_in[13];
  const float* sem_w    = (const float*)d_in[14];
  const float* sem_b    = (const float*)d_in[15];
  const float* str_w    = (const float*)d_in[16];
  const float* str_b    = (const float*)d_in[17];
  const float* g1_w     = (const float*)d_in[18];
  const float* g1_b     = (const float*)d_in[19];
  const float* g2_w     = (const float*)d_in[20];
  const float* g2_b     = (const float*)d_in[21];
  const float* rel_w    = (const float*)d_in[22];   // [2,8,128,128]
  const float* root_w   = (const float*)d_in[23];   // [2,128,128]
  const float* rg_bias  = (const float*)d_in[24];   // [2,128]
  const float* lora_a   = (const float*)d_in[25];   // [2,128,8]
  const float* lora_b   = (const float*)d_in[26];   // [2,8,128]
  const int*   e_idx    = (const int*)d_in[27];     // [2,E]
  const int*   e_type   = (const int*)d_in[28];
  const int*   batch    = (const int*)d_in[29];
  const int*   dom_ids  = (const int*)d_in[30];
  const int*   e_pid    = (const int*)d_in[31];

  const int N = in_sizes[29];
  const int E = in_sizes[28];
  const int H = 128;
  const int* src = e_idx;
  const int* dst = e_idx + E;

  float* ws = (float*)d_ws;
  size_t NH = (size_t)N * H;
  float* h    = ws;
  float* pc   = ws + 1 * NH;
  float* prop = ws + 2 * NH;
  float* curr = ws + 3 * NH;
  float* cin  = ws + 4 * NH;
  float* zr   = ws + 5 * NH;
  float* conv = ws + 6 * NH;
  float* cntp = ws + 7 * NH;                 // N*16
  float* cnt  = cntp + (size_t)N * 16;       // N*8
  float* weff = cnt + (size_t)N * 8;         // 128*128

  const int total = N * H;
  dim3 gblk((N + 31) / 32);                  // GEMM row-tile grid
  dim3 eblk((total + 255) / 256);            // elementwise grid

  // 1-2) text/path projections (K=768)
  gemm_kernel<768, false><<<gblk, 256, 0, stream>>>(x_text, nullptr, text_w, text_b, h, N, 0);
  gemm_kernel<768, false><<<gblk, 256, 0, stream>>>(x_path, nullptr, path_w, path_b, pc, N, 0);
  // 3) domain modulation
  modulate_kernel<<<eblk, 256, 0, stream>>>(h, pc, dom_emb, dom_ids, batch, total);
  // 4) path fusion (in-place on pc: each block finishes reading its rows before storing)
  gemm_kernel<128, false><<<gblk, 256, 0, stream>>>(pc, nullptr, fus_w, fus_b, pc, N, 0);
  // 5) path gate MLP
  gemm_kernel<256, true ><<<gblk, 256, 0, stream>>>(h, pc, pg1_w, pg1_b, zr, N, 1);
  gemm_kernel<128, false><<<gblk, 256, 0, stream>>>(zr, nullptr, pg2_w, pg2_b, conv, N, 2);
  // 6) edge property aggregation (counts trick, P=16)
  (void)hipMemsetAsync(cntp, 0, (size_t)N * 24 * sizeof(float), stream);
  edge_counts_kernel<<<(E + 255) / 256, 256, 0, stream>>>(src, dst, e_type, e_pid, cntp, cnt, E);
  prop_kernel<<<N, 128, 0, stream>>>(cntp, prop_emb, prop);
  // 7) gated fusion + property prompts: h = pg*h + (1-pg)*path_ctx + prop_agg
  combine1_kernel<<<eblk, 256, 0, stream>>>(h, pc, conv, prop, total);

  // 8) RGCN layers
  for (int l = 0; l < 2; ++l) {
    weff_kernel<<<64, 256, 0, stream>>>(root_w + (size_t)l * 128 * 128,
                                        lora_a + (size_t)l * 128 * 8,
                                        lora_b + (size_t)l * 8 * 128, weff);
    add_kernel<<<eblk, 256, 0, stream>>>(cin, (l == 0 ? h : curr), prop, total);
    // conv = cin @ (root + s*A@B) + bias
    gemm_kernel<128, false><<<gblk, 256, 0, stream>>>(cin, nullptr, weff, rg_bias + l * 128, conv, N, 0);
    // + per-relation mean-aggregated messages (transform-then-scatter)
    for (int r = 0; r < 8; ++r) {
      gemm_kernel<128, false><<<gblk, 256, 0, stream>>>(
          cin, nullptr, rel_w + ((size_t)(l * 8 + r)) * 128 * 128, nullptr, zr, N, 0);
      scatter_rel_kernel<<<(E + 7) / 8, 256, 0, stream>>>(zr, src, dst, e_type, cnt, conv, E, r);
    }
    relu_kernel<<<eblk, 256, 0, stream>>>(curr, conv, total);
  }

  // 9) semantic/structural gated fusion
  gemm_kernel<128, false><<<gblk, 256, 0, stream>>>(h, nullptr, sem_w, sem_b, pc, N, 0);    // sem
  gemm_kernel<128, false><<<gblk, 256, 0, stream>>>(curr, nullptr, str_w, str_b, cin, N, 0);// stru
  gemm_kernel<256, true ><<<gblk, 256, 0, stream>>>(pc, cin, g1_w, g1_b, zr, N, 1);
  gemm_kernel<128, false><<<gblk, 256, 0, stream>>>(zr, nullptr, g2_w, g2_b, conv, N, 2);   // fg
  final_kernel<<<eblk, 256, 0, stream>>>((float*)d_out, conv, pc, cin, total);
}